// MelDecoder_19593640805077
// MI455X (gfx1250) — compile-verified
//
#include <hip/hip_runtime.h>

// ---------------------------------------------------------------------------
// Glottal pulse synthesis, MI455X (gfx1250, wave32).
//  prep:  one block per batch row. Wave 0 does the frame-level exclusive
//         scan (breaks the 960k-sample serial cumsum down to T=4000 frames,
//         since the upsampled increment is constant within a frame);
//         all 8 waves then emit two packed float4 param records per frame.
//  synth: 1 thread = 8 samples (one frame). Two async b128 global->LDS DMAs
//         fetch the noise while the sin/log2/exp2 pulse chain runs, then
//         s_wait_asynccnt, modulate, two b128 stores.
// ---------------------------------------------------------------------------

#define B_    8
#define T_    4000
#define HOP_  240
#define NFRAME (B_ * T_)            // 32000
#define NSAMP  (B_ * T_ * HOP_)     // 7,680,000

#define SR_F          24000.0f
#define TWO_PI_F      6.28318530717958647692f
#define INV_TWO_PI_F  0.15915494309189533577f
#define PI_F          3.14159265358979323846f

#define SYNTH_BLOCK 256
#define VEC 8                        // 240 % 8 == 0 -> one frame per thread
#define TILE (SYNTH_BLOCK * VEC)     // 2048 samples / block -> 3750 blocks

#if defined(__AMDGCN__) && \
    __has_builtin(__builtin_amdgcn_global_load_async_to_lds_b128) && \
    __has_builtin(__builtin_amdgcn_s_wait_asynccnt)
#define HAS_ASYNC 1
#else
#define HAS_ASYNC 0
#endif

#if defined(__AMDGCN__) && __has_builtin(__builtin_amdgcn_exp2f)
#define EXP2F(x) __builtin_amdgcn_exp2f(x)      // v_exp_f32
#else
#define EXP2F(x) exp2f(x)
#endif

// pointer types matching the async builtin's parameters (probe-derived)
typedef int v4i __attribute__((vector_size(16)));
typedef __attribute__((address_space(3))) v4i lds_v4i;

__device__ __forceinline__ float fast_sigmoid(float x) {
    return 1.0f / (1.0f + __expf(-x));   // v_exp_f32 path
}

// ---------------------------------------------------------------------------
// prep: scan + packed params.
//   p0[bt] = { frame_base, inc, pi/oq, oq }
//   p1[bt] = { 1/(1-oq), -oq/(1-oq), cf, sh }
// ---------------------------------------------------------------------------
__global__ void __launch_bounds__(256)
prep_kernel(const float* __restrict__ f0,
            const float* __restrict__ gp,          // [B,3,T]
            float4* __restrict__ p0_w,
            float4* __restrict__ p1_w)
{
    __shared__ float lds_fb[T_];                   // 16 KB: frame_base scratch
    const int b   = blockIdx.x;
    const int tid = threadIdx.x;

    if (tid < 32) {                                // single wave: frame scan
        const int  lane = tid;
        const int  SEG  = T_ / 32;                 // 125
        const int  t0   = lane * SEG;
        const float C1  = TWO_PI_F * (float)HOP_ / SR_F;   // 240*2pi/24000

        float s = 0.0f;
        for (int i = 0; i < SEG; ++i) s += f0[b * T_ + t0 + i] * C1;

        float v = s;                               // inclusive wave32 scan
        #pragma unroll
        for (int d = 1; d < 32; d <<= 1) {
            float n = __shfl_up(v, d, 32);
            if (lane >= d) v += n;
        }
        float run = v - s;                         // exclusive prefix
        for (int i = 0; i < SEG; ++i) {
            lds_fb[t0 + i] = run;
            run += f0[b * T_ + t0 + i] * C1;
        }
    }
    __syncthreads();

    for (int t = tid; t < T_; t += 256) {
        float g0 = gp[(b * 3 + 0) * T_ + t];
        float g1 = gp[(b * 3 + 1) * T_ + t];
        float g2 = gp[(b * 3 + 2) * T_ + t];
        float oq   = fast_sigmoid(g0) * 0.5f + 0.25f;      // [0.25, 0.75]
        float tilt = fast_sigmoid(g1) * 0.5f;
        float cf   = (1.0f - tilt) * 1.5f + 0.5f;          // [1.25, 2.0]
        float sh   = fast_sigmoid(g2) * 0.05f;

        float fb   = lds_fb[t];
        float inc  = f0[b * T_ + t] * (TWO_PI_F / SR_F);
        float a    = 1.0f / (1.0f - oq);                   // closing ramp slope
        float bb   = -oq * a;                              // closing ramp bias
        float pioq = PI_F / oq;                            // open-phase arg scale

        p0_w[b * T_ + t] = make_float4(fb, inc, pioq, oq);
        p1_w[b * T_ + t] = make_float4(a, bb, cf, sh);
    }
}

// ---------------------------------------------------------------------------
// synth: 8 samples / thread, all within one frame.
// ---------------------------------------------------------------------------
__global__ void __launch_bounds__(SYNTH_BLOCK)
synth_kernel(const float* __restrict__ noise,
             const float4* __restrict__ p0_w,
             const float4* __restrict__ p1_w,
             float* __restrict__ out)
{
    __shared__ __align__(16) float lbuf[TILE];     // 8 KB
    const int tid = threadIdx.x;
    const int i0  = blockIdx.x * TILE + tid * VEC; // first flat sample

#if HAS_ASYNC
    // fire-and-forget 32B of noise -> LDS (ASYNCcnt); overlaps the
    // transcendental pulse computation below.
    __builtin_amdgcn_global_load_async_to_lds_b128(
        (v4i*)(void*)(noise + i0),
        (lds_v4i*)&lbuf[tid * VEC], 0, 0);
    __builtin_amdgcn_global_load_async_to_lds_b128(
        (v4i*)(void*)(noise + i0 + 4),
        (lds_v4i*)&lbuf[tid * VEC + 4], 0, 0);
#else
    (void)lbuf;
#endif

    // flat layout [B, T*HOP]: i/HOP == b*T + t directly
    const int bt = i0 / HOP_;
    const int k0 = i0 - bt * HOP_;

    const float4 p0 = p0_w[bt];    // { fb, inc, pi/oq, oq }
    const float4 p1 = p1_w[bt];    // { a, b, cf, sh }
    const float fb = p0.x, inc = p0.y, pioq = p0.z, oq = p0.w;
    const float a  = p1.x, bb  = p1.y, cf   = p1.z, sh = p1.w;

    float pulse[VEC];
    #pragma unroll
    for (int j = 0; j < VEC; ++j) {
        float phase = fb + (float)(k0 + j) * inc;
        float cyc   = phase * INV_TWO_PI_F;
        float tn    = cyc - floorf(cyc);                 // t_norm in [0,1)

        // open phase: sin(pi * tn / oq)                 (v_sin_f32)
        float po = __sinf(tn * pioq);

        // closing phase: 1 - tc^cf with 0^cf == 0       (v_log_f32 + v_exp_f32)
        float tc = fmaf(tn, a, bb);                      // (tn - oq)/(1 - oq)
        tc = fminf(fmaxf(tc, 0.0f), 1.0f);
        float pw = EXP2F(cf * __log2f(fmaxf(tc, 1e-30f)));
        pw = (tc <= 0.0f) ? 0.0f : pw;
        float pc = 1.0f - pw;

        // tn < 1 always, so close_mask == !open_mask
        pulse[j] = (tn < oq) ? po : pc;
    }

#if HAS_ASYNC
    __builtin_amdgcn_s_wait_asynccnt(0);                 // DMA complete
#endif

    #pragma unroll
    for (int h = 0; h < VEC / 4; ++h) {
        float4 nz;
#if HAS_ASYNC
        nz = *(const float4*)&lbuf[tid * VEC + h * 4];   // ds_load_b128
#else
        nz = *(const float4*)(noise + i0 + h * 4);
#endif
        float4 o;
        o.x = pulse[h * 4 + 0] * fmaf(sh, nz.x - 0.5f, 1.0f);
        o.y = pulse[h * 4 + 1] * fmaf(sh, nz.y - 0.5f, 1.0f);
        o.z = pulse[h * 4 + 2] * fmaf(sh, nz.z - 0.5f, 1.0f);
        o.w = pulse[h * 4 + 3] * fmaf(sh, nz.w - 0.5f, 1.0f);
        *(float4*)(out + i0 + h * 4) = o;                // global_store_b128
    }
}

// ---------------------------------------------------------------------------
extern "C" void kernel_launch(void* const* d_in, const int* in_sizes, int n_in,
                              void* d_out, int out_size, void* d_ws, size_t ws_size,
                              hipStream_t stream)
{
    const float* f0    = (const float*)d_in[0];   // [8, 4000]
    const float* gp    = (const float*)d_in[1];   // [8, 3, 4000]
    const float* noise = (const float*)d_in[2];   // [8, 960000]
    float* out = (float*)d_out;                   // [8, 960000]

    // workspace: two packed float4 param arrays, 2 * 32000 * 16B = 1 MB
    float4* p0_w = (float4*)d_ws;
    float4* p1_w = p0_w + NFRAME;

    prep_kernel<<<B_, 256, 0, stream>>>(f0, gp, p0_w, p1_w);

    synth_kernel<<<NSAMP / TILE, SYNTH_BLOCK, 0, stream>>>(
        noise, p0_w, p1_w, out);
}